// GCNLSTMModel_67190468378873
// MI455X (gfx1250) — compile-verified
//
#include <hip/hip_runtime.h>

// ---------------------------------------------------------------------------
// GCN message-passing layer for MI455X (gfx1250, wave32).
// Memory-bound problem (~440MB HBM traffic vs <1us of fp32-WMMA compute),
// so: fp32 V_WMMA_F32_16X16X4_F32 for exactness, LDS-tiled GEMM with 128-bit
// async global->LDS copies (GLOBAL_LOAD_ASYNC_TO_LDS_B128, ASYNCcnt),
// L2-friendly grid order, atomic-free deterministic gather.
// ---------------------------------------------------------------------------

#define DIM    4096
#define NCOLS  12288      // 3*DIM
#define NREG   2048       // N_IMG*REG
#define NIMG   64
#define NUMREL 20
#define RPI    32
#define BM     128
#define BN     64
#define BK     32
#define LDA    36         // padded LDS stride (floats) for the A tile

typedef __attribute__((ext_vector_type(2))) float v2f;
typedef __attribute__((ext_vector_type(8))) float v8f;
typedef __attribute__((ext_vector_type(4))) int   v4i;

// All AMDGPU builtins are referenced ONLY inside __HIP_DEVICE_COMPILE__
// blocks so the x86 host pass never has to resolve them.

// Async global->LDS copy (CDNA5 GLOBAL_LOAD_ASYNC_TO_LDS_B128, ASYNCcnt).
// Device-pass probe (round 1) showed signature (int4 AS1*, int4 AS3*, imm, imm).
#if defined(__HIP_DEVICE_COMPILE__) && \
    __has_builtin(__builtin_amdgcn_global_load_async_to_lds_b128) && \
    __has_builtin(__builtin_amdgcn_s_wait_asynccnt)
#define USE_ASYNC_COPY 1
__device__ __forceinline__ void copy16(const float* g, float* l) {
  __builtin_amdgcn_global_load_async_to_lds_b128(
      (__attribute__((address_space(1))) v4i*)g,
      (__attribute__((address_space(3))) v4i*)l,
      0, 0);
}
__device__ __forceinline__ void copy_fence() {
  __builtin_amdgcn_s_wait_asynccnt(0);
}
#else
#define USE_ASYNC_COPY 0
__device__ __forceinline__ void copy16(const float* g, float* l) {
  *(float4*)l = *(const float4*)g;
}
__device__ __forceinline__ void copy_fence() {}
#endif

// fp32 WMMA wrapper: builtin only in the device pass.
__device__ __forceinline__ v8f wmma_f32_16x16x4(v2f a, v2f b, v8f c) {
#if defined(__HIP_DEVICE_COMPILE__)
  return __builtin_amdgcn_wmma_f32_16x16x4_f32(false, a, false, b,
                                               (short)0, c, false, false);
#else
  return c;
#endif
}

// ---------------------------------------------------------------------------
// Kernel 1: neighbor_feats = region_feats @ W_conv   (2048x4096 @ 4096x12288)
// Block = 256 threads = 8 waves arranged 4(M) x 2(N); each wave owns a 32x32
// output tile = 2x2 WMMA fragments. grid.x = M tiles (fastest) so the 16
// blocks sharing one W_conv column slab are dispatch-adjacent -> L2 reuse.
// ---------------------------------------------------------------------------
extern "C" __global__ __launch_bounds__(256)
void nf_gemm_wmma(const float* __restrict__ A,   // region_feats [NREG][DIM]
                  const float* __restrict__ Bw,  // W_conv       [DIM][NCOLS]
                  float* __restrict__ C) {       // neighbor     [NREG][NCOLS]
  __shared__ float Asm[BM * LDA];
  __shared__ float Bsm[BK * BN];

  const int tid  = threadIdx.x;
  const int lane = tid & 31;
  const int wave = tid >> 5;
  const int wm   = wave >> 1;       // 0..3
  const int wn   = wave & 1;        // 0..1
  const int lm   = lane & 15;       // M (or N) index inside a 16-wide frag
  const int kh   = lane >> 4;       // K half-select (0/1)
  const int blockM = blockIdx.x * BM;
  const int blockN = blockIdx.y * BN;

  v8f acc00 = {}, acc01 = {}, acc10 = {}, acc11 = {};

  for (int k0 = 0; k0 < DIM; k0 += BK) {
    __syncthreads();  // previous compute done before LDS overwrite
    // A tile: 128x32 f32 = 1024 float4, 4 per thread
#pragma unroll
    for (int r = 0; r < 4; ++r) {
      int i = tid + 256 * r;
      int row = i >> 3, c4 = (i & 7) << 2;
      copy16(&A[(size_t)(blockM + row) * DIM + k0 + c4], &Asm[row * LDA + c4]);
    }
    // B tile: 32x64 f32 = 512 float4, 2 per thread
#pragma unroll
    for (int r = 0; r < 2; ++r) {
      int i = tid + 256 * r;
      int row = i >> 4, c4 = (i & 15) << 2;
      copy16(&Bw[(size_t)(k0 + row) * NCOLS + blockN + c4], &Bsm[row * BN + c4]);
    }
    copy_fence();
    __syncthreads();

#pragma unroll
    for (int kk = 0; kk < BK; kk += 4) {
      const int ka = kk + 2 * kh;   // per-lane K base (ISA 16x4 f32 layout)
      const float* ap0 = &Asm[(wm * 32 + lm) * LDA + ka];
      const float* ap1 = ap0 + 16 * LDA;
      v2f a0 = {ap0[0], ap0[1]};
      v2f a1 = {ap1[0], ap1[1]};
      const float* bp0 = &Bsm[ka * BN + wn * 32 + lm];
      v2f b0 = {bp0[0],  bp0[BN]};
      v2f b1 = {bp0[16], bp0[16 + BN]};
      acc00 = wmma_f32_16x16x4(a0, b0, acc00);
      acc01 = wmma_f32_16x16x4(a0, b1, acc01);
      acc10 = wmma_f32_16x16x4(a1, b0, acc10);
      acc11 = wmma_f32_16x16x4(a1, b1, acc11);
    }
  }

  // D layout: VGPR v <-> M = v + 8*(lane/16); lane%16 <-> N
  const int colBase = blockN + wn * 32 + lm;
  const int rowBase = blockM + wm * 32 + 8 * kh;
#pragma unroll
  for (int v = 0; v < 8; ++v) {
    C[(size_t)(rowBase + v)      * NCOLS + colBase]      = acc00[v];
    C[(size_t)(rowBase + v)      * NCOLS + colBase + 16] = acc01[v];
    C[(size_t)(rowBase + 16 + v) * NCOLS + colBase]      = acc10[v];
    C[(size_t)(rowBase + 16 + v) * NCOLS + colBase + 16] = acc11[v];
  }
}

// ---------------------------------------------------------------------------
// Kernel 2: gate_feats = region_feats @ W_g  (2048x4096 @ 4096x3), pre-sigmoid
// ---------------------------------------------------------------------------
extern "C" __global__ __launch_bounds__(128)
void gate_gemv(const float* __restrict__ RF, const float* __restrict__ Wg,
               float* __restrict__ gate_out) {
  const int row = blockIdx.x;   // 0..2047
  const int t   = threadIdx.x;  // 0..127
  float s0 = 0.f, s1 = 0.f, s2 = 0.f;
  for (int d = t; d < DIM; d += 128) {
    float r = RF[(size_t)row * DIM + d];
    s0 += r * Wg[d * 3 + 0];
    s1 += r * Wg[d * 3 + 1];
    s2 += r * Wg[d * 3 + 2];
  }
  __shared__ float red[3][128];
  red[0][t] = s0; red[1][t] = s1; red[2][t] = s2;
  __syncthreads();
  for (int off = 64; off > 0; off >>= 1) {
    if (t < off) {
      red[0][t] += red[0][t + off];
      red[1][t] += red[1][t + off];
      red[2][t] += red[2][t + off];
    }
    __syncthreads();
  }
  if (t == 0) {
    gate_out[row * 3 + 0] = red[0][0];
    gate_out[row * 3 + 1] = red[1][0];
    gate_out[row * 3 + 2] = red[2][0];
  }
}

// ---------------------------------------------------------------------------
// Kernel 3: gather + gate + bias + ReLU. One block per target region; scan
// only that image's 20 edges (deterministic order, no atomics).
//   out[t] = relu( g_self*(nf[3t+2]+b_lab[0])
//                + sum_{e: s_e==t} g*(nf[3o_e+0]+b_lab[p_e])
//                + sum_{e: o_e==t} g*(nf[3s_e+1]+b_lab[p_e]) )
// ---------------------------------------------------------------------------
extern "C" __global__ __launch_bounds__(256)
void msg_gather(const float* __restrict__ nf, const float* __restrict__ gatef,
                const int* __restrict__ rels, const int* __restrict__ preds,
                const float* __restrict__ b_lab, const float* __restrict__ b_glab,
                float* __restrict__ out) {
  const int t   = blockIdx.x;   // target region 0..2047
  const int img = t >> 5;       // /REG

  __shared__ int   m_src[64];
  __shared__ int   m_pred[64];
  __shared__ float m_gate[64];
  __shared__ int   m_cnt;

  if (threadIdx.x == 0) {
    int cnt = 0;
    m_src[cnt] = t * 3 + 2; m_pred[cnt] = 0; ++cnt;          // self loop
    for (int e = 0; e < NUMREL; ++e) {                       // t as subject
      int rr = (img * RPI + e) * 3;
      if (rels[rr + 1] == t) {
        m_src[cnt] = rels[rr + 2] * 3 + 0;
        m_pred[cnt] = preds[img * RPI + e];
        ++cnt;
      }
    }
    for (int e = 0; e < NUMREL; ++e) {                       // t as object
      int rr = (img * RPI + e) * 3;
      if (rels[rr + 2] == t) {
        m_src[cnt] = rels[rr + 1] * 3 + 1;
        m_pred[cnt] = preds[img * RPI + e];
        ++cnt;
      }
    }
    m_cnt = cnt;
  }
  __syncthreads();

  const int cnt = m_cnt;
  if (threadIdx.x < cnt) {
    float g = gatef[m_src[threadIdx.x]] + b_glab[m_pred[threadIdx.x]];
    m_gate[threadIdx.x] = 1.0f / (1.0f + expf(-g));
  }
  __syncthreads();

  for (int d = threadIdx.x; d < DIM; d += 256) {
    float acc = 0.f;
    for (int m = 0; m < cnt; ++m) {
      acc += m_gate[m] * (nf[(size_t)m_src[m] * DIM + d] +
                          b_lab[(size_t)m_pred[m] * DIM + d]);
    }
    out[(size_t)t * DIM + d] = fmaxf(acc, 0.f);
  }
}

// ---------------------------------------------------------------------------
extern "C" void kernel_launch(void* const* d_in, const int* in_sizes, int n_in,
                              void* d_out, int out_size, void* d_ws, size_t ws_size,
                              hipStream_t stream) {
  (void)in_sizes; (void)n_in; (void)out_size; (void)ws_size;
  // setup_inputs order:
  // 0 im_inds 1 region_feats 2 pred_classes 3 rels 4 obj_classes
  // 5 seq_labels 6 mask_labels 7 W_conv 8 W_g 9 b_lab 10 b_glab
  const float* region_feats = (const float*)d_in[1];
  const int*   pred_classes = (const int*)d_in[2];
  const int*   rels         = (const int*)d_in[3];
  const float* W_conv       = (const float*)d_in[7];
  const float* W_g          = (const float*)d_in[8];
  const float* b_lab        = (const float*)d_in[9];
  const float* b_glab       = (const float*)d_in[10];
  float* out = (float*)d_out;

  float* nf    = (float*)d_ws;                     // [3*NREG][DIM] = 100.7 MB
  float* gatef = nf + (size_t)3 * NREG * DIM;      // [3*NREG]

  dim3 ggrid(NREG / BM, NCOLS / BN);               // (16, 192): M fastest
  nf_gemm_wmma<<<ggrid, 256, 0, stream>>>(region_feats, W_conv, nf);
  gate_gemv<<<NREG, 128, 0, stream>>>(region_feats, W_g, gatef);
  msg_gather<<<NREG, 256, 0, stream>>>(nf, gatef, rels, pred_classes,
                                       b_lab, b_glab, out);
}